// DTWLayer_33406255628870
// MI455X (gfx1250) — compile-verified
//
#include <hip/hip_runtime.h>

typedef __attribute__((ext_vector_type(2))) float v2f;
typedef __attribute__((ext_vector_type(8))) float v8f;

constexpr int B_ = 16, N_ = 1024, M_ = 4096, D_ = 128;
constexpr int WPT = M_ / 256;   // 16 columns per DP thread

// -------------------- row norms: x2[b,i], y2[b,j] --------------------
__global__ __launch_bounds__(128)
void row_norms_kernel(const float* __restrict__ x, const float* __restrict__ y,
                      float* __restrict__ x2, float* __restrict__ y2) {
    int gid = blockIdx.x * 128 + threadIdx.x;
    int total_x = B_ * N_;
    int total = total_x + B_ * M_;
    if (gid >= total) return;
    const float* base = (gid < total_x) ? (x + (size_t)gid * D_)
                                        : (y + (size_t)(gid - total_x) * D_);
    float s = 0.f;
    for (int k = 0; k < D_; k += 4) {
        float4 v = *(const float4*)(base + k);
        s += v.x * v.x + v.y * v.y + v.z * v.z + v.w * v.w;
    }
    if (gid < total_x) x2[gid] = s; else y2[gid - total_x] = s;
}

// -------------------- WMMA f32 GEMM: d = x2 + y2 - 2*x@y^T --------------------
// block = 128 threads (4 waves), tile = 32 rows x 64 cols, K=128 staged in LDS
__global__ __launch_bounds__(128)
void sqdist_wmma_kernel(const float* __restrict__ x, const float* __restrict__ y,
                        const float* __restrict__ x2, const float* __restrict__ y2,
                        float* __restrict__ dmat) {
    __shared__ float xs[32 * 132];   // +4 pad: stride 132 floats avoids bank conflicts
    __shared__ float ys[64 * 132];
    int b = blockIdx.z;
    int rowBase = blockIdx.y * 32;
    int colBase = blockIdx.x * 64;
    int tid = threadIdx.x;

    const float* xsrc = x + ((size_t)b * N_ + rowBase) * D_;
    for (int idx = tid; idx < 32 * 32; idx += 128) {          // 32 rows x 32 float4
        int r = idx >> 5, c4 = idx & 31;
        float4 v = *(const float4*)(xsrc + (size_t)r * D_ + c4 * 4);
        *(float4*)(&xs[r * 132 + c4 * 4]) = v;
    }
    const float* ysrc = y + ((size_t)b * M_ + colBase) * D_;
    for (int idx = tid; idx < 64 * 32; idx += 128) {          // 64 rows x 32 float4
        int r = idx >> 5, c4 = idx & 31;
        float4 v = *(const float4*)(ysrc + (size_t)r * D_ + c4 * 4);
        *(float4*)(&ys[r * 132 + c4 * 4]) = v;
    }
    __syncthreads();

    int wid = tid >> 5, lane = tid & 31;
    int wr = (wid >> 1) * 16;          // wave's 16-row slice within 32
    int wc = (wid & 1) * 32;           // wave's 32-col slice within 64
    int mrow = lane & 15;
    int kh = (lane >> 4) * 2;          // lanes 16-31 hold K = k+2, k+3

    v8f acc0 = {}; v8f acc1 = {};
    const float* xa  = &xs[(wr + mrow) * 132];
    const float* yb0 = &ys[(wc + mrow) * 132];
    const float* yb1 = &ys[(wc + 16 + mrow) * 132];
    for (int k = 0; k < D_; k += 4) {
        v2f a, b0, b1;
        a[0]  = xa[k + kh];  a[1]  = xa[k + kh + 1];
        b0[0] = yb0[k + kh]; b0[1] = yb0[k + kh + 1];
        b1[0] = yb1[k + kh]; b1[1] = yb1[k + kh + 1];
        acc0 = __builtin_amdgcn_wmma_f32_16x16x4_f32(false, a, false, b0,
                                                     (short)0, acc0, false, false);
        acc1 = __builtin_amdgcn_wmma_f32_16x16x4_f32(false, a, false, b1,
                                                     (short)0, acc1, false, false);
    }

    // epilogue: D layout -> VGPR r holds (M = r + 8*(lane>>4), Nc = lane&15)
    int col0 = colBase + wc + (lane & 15);
    float y2c0 = y2[b * M_ + col0];
    float y2c1 = y2[b * M_ + col0 + 16];
    int rsel = (lane >> 4) * 8;
    float* dout = dmat + (size_t)b * N_ * M_;
#pragma unroll
    for (int r = 0; r < 8; r++) {
        int row = rowBase + wr + r + rsel;
        float xn = x2[b * N_ + row];
        dout[(size_t)row * M_ + col0]      = xn + y2c0 - 2.0f * acc0[r];
        dout[(size_t)row * M_ + col0 + 16] = xn + y2c1 - 2.0f * acc1[r];
    }
}

// -------------------- DP: one block per batch, cumsum/cummin row scan --------------------
__global__ __launch_bounds__(256)
void dtw_dp_kernel(const float* __restrict__ dmat, unsigned int* __restrict__ choice,
                   int* __restrict__ startj) {
    __shared__ float prevRow[M_];     // 16 KB rolling D row
    __shared__ float red[256];
    __shared__ float edge[256];
    __shared__ int   redi[256];
    const float INF = __builtin_inff();
    int b = blockIdx.x;
    int t = threadIdx.x;
    int j0 = t * WPT;

    // row 0: D[0,:] = d[0,:]
    {
        const float* drow0 = dmat + (size_t)b * N_ * M_;
#pragma unroll
        for (int q = 0; q < WPT / 4; q++) {
            float4 v = *(const float4*)(drow0 + j0 + q * 4);
            prevRow[j0 + q * 4 + 0] = v.x; prevRow[j0 + q * 4 + 1] = v.y;
            prevRow[j0 + q * 4 + 2] = v.z; prevRow[j0 + q * 4 + 3] = v.w;
        }
        // prefetch row 1 (global_prefetch_b8): hide HBM latency behind the scans
        __builtin_prefetch(drow0 + M_ + j0, 0, 1);
    }
    __syncthreads();

    for (int i = 1; i < N_; i++) {
        const float* drow = dmat + ((size_t)b * N_ + i) * M_;
        float dl[WPT];
#pragma unroll
        for (int q = 0; q < WPT / 4; q++) {
            float4 v = *(const float4*)(drow + j0 + q * 4);
            dl[q * 4 + 0] = v.x; dl[q * 4 + 1] = v.y;
            dl[q * 4 + 2] = v.z; dl[q * 4 + 3] = v.w;
        }
        // prefetch next row's 64B chunk per thread while this row's scans run
        if (i + 1 < N_) __builtin_prefetch(drow + M_ + j0, 0, 1);
        // local inclusive cumsum
        float cs[WPT]; float s = 0.f;
#pragma unroll
        for (int l = 0; l < WPT; l++) { s += dl[l]; cs[l] = s; }
        // block inclusive sum scan of per-thread totals
        red[t] = s; __syncthreads();
        for (int off = 1; off < 256; off <<= 1) {
            float mine = red[t];
            float oth  = (t >= off) ? red[t - off] : 0.f;
            __syncthreads();
            red[t] = mine + oth;
            __syncthreads();
        }
        float offsum = (t > 0) ? red[t - 1] : 0.f;
        float c[WPT];
#pragma unroll
        for (int l = 0; l < WPT; l++) c[l] = offsum + cs[l];
        // g[j] = min(prev[j], prev[j-1]) - c[j-1]
        float g[WPT];
#pragma unroll
        for (int l = 0; l < WPT; l++) {
            int j = j0 + l;
            float up   = prevRow[j];
            float diag = (j > 0) ? prevRow[j - 1] : INF;
            float m = fminf(up, diag);
            float csh = (l > 0) ? (offsum + cs[l - 1]) : ((t > 0) ? offsum : 0.f);
            g[l] = m - csh;
        }
        // local inclusive cummin
        float gm[WPT]; float mn = INF;
#pragma unroll
        for (int l = 0; l < WPT; l++) { mn = fminf(mn, g[l]); gm[l] = mn; }
        // block inclusive min scan
        red[t] = mn; __syncthreads();
        for (int off = 1; off < 256; off <<= 1) {
            float mine = red[t];
            float oth  = (t >= off) ? red[t - off] : INF;
            __syncthreads();
            red[t] = fminf(mine, oth);
            __syncthreads();
        }
        float offmin = (t > 0) ? red[t - 1] : INF;
        float rowv[WPT];
#pragma unroll
        for (int l = 0; l < WPT; l++) rowv[l] = fminf(offmin, gm[l]) + c[l];

        edge[t] = rowv[WPT - 1];
        __syncthreads();
        float leftFirst = (t > 0) ? edge[t - 1] : INF;

        // record backtrack choices (0=diag,1=up,2=left; first-min tie order)
        unsigned int pack = 0;
#pragma unroll
        for (int l = 0; l < WPT; l++) {
            int j = j0 + l;
            float diag = (j > 0) ? prevRow[j - 1] : INF;
            float up   = prevRow[j];
            float left = (l > 0) ? rowv[l - 1] : leftFirst;   // j==0 -> leftFirst==INF
            unsigned int best = 0; float bv = diag;
            if (up   < bv) { best = 1; bv = up; }
            if (left < bv) { best = 2; bv = left; }
            pack |= best << (2 * l);
        }
        choice[((size_t)b * N_ + i) * 256 + t] = pack;
        __syncthreads();                // everyone done reading prevRow
#pragma unroll
        for (int l = 0; l < WPT; l++) prevRow[j0 + l] = rowv[l];
        __syncthreads();
    }

    // argmin of last row (first minimum wins, matching jnp.argmin)
    float bv = prevRow[j0]; int bi = j0;
#pragma unroll
    for (int l = 1; l < WPT; l++) {
        float v = prevRow[j0 + l];
        if (v < bv) { bv = v; bi = j0 + l; }
    }
    red[t] = bv; redi[t] = bi; __syncthreads();
    for (int off = 128; off > 0; off >>= 1) {
        if (t < off) {
            float v2 = red[t + off]; int i2 = redi[t + off];
            if (v2 < red[t] || (v2 == red[t] && i2 < redi[t])) { red[t] = v2; redi[t] = i2; }
        }
        __syncthreads();
    }
    if (t == 0) startj[b] = redi[0];
}

// -------------------- backtrack + outputs --------------------
__global__ __launch_bounds__(128)
void dtw_backtrack_kernel(const float* __restrict__ dmat, const unsigned int* __restrict__ choice,
                          const int* __restrict__ startj, const float* __restrict__ x_t,
                          const float* __restrict__ y_t, float* __restrict__ out) {
    int b = blockIdx.x;
    int tid = threadIdx.x;
    float* out_cost = out;
    float* out_ts   = out + B_;
    float* out_vs   = out + B_ + B_ * N_;
    for (int i = tid; i < N_; i += blockDim.x)
        out_ts[b * N_ + i] = x_t[b * N_ + i];
    if (tid != 0) return;

    int i = N_ - 1, j = startj[b];
    float cost = 0.f;
    out_vs[b * N_ + i] = y_t[b * M_ + j];      // arrival j = max j for this row
    for (int step = 0; step < N_ + M_; step++) {
        cost += dmat[((size_t)b * N_ + i) * M_ + j];
        if (i == 0) break;
        unsigned int w = choice[((size_t)b * N_ + i) * 256 + (j >> 4)];
        unsigned int c2 = (w >> ((j & 15) * 2)) & 3u;
        if (c2 == 0)      { i--; j--; }
        else if (c2 == 1) { i--; }
        else              { j--; continue; }   // left move: same row, no record
        out_vs[b * N_ + i] = y_t[b * M_ + j];
    }
    out_cost[b] = cost;
}

// -------------------- launcher --------------------
extern "C" void kernel_launch(void* const* d_in, const int* in_sizes, int n_in,
                              void* d_out, int out_size, void* d_ws, size_t ws_size,
                              hipStream_t stream) {
    (void)in_sizes; (void)n_in; (void)out_size; (void)ws_size;
    const float* x   = (const float*)d_in[0];   // [B,N,D]
    const float* y   = (const float*)d_in[1];   // [B,M,D]
    const float* x_t = (const float*)d_in[2];   // [B,N]
    const float* y_t = (const float*)d_in[3];   // [B,M]
    float* out = (float*)d_out;                 // [B] + [B,N] + [B,N]

    char* ws = (char*)d_ws;
    size_t off = 0;
    float* dmat = (float*)(ws + off);          off += (size_t)B_ * N_ * M_ * sizeof(float);   // 256 MB
    float* x2   = (float*)(ws + off);          off += (size_t)B_ * N_ * sizeof(float);
    float* y2   = (float*)(ws + off);          off += (size_t)B_ * M_ * sizeof(float);
    unsigned int* choice = (unsigned int*)(ws + off); off += (size_t)B_ * N_ * 256 * sizeof(unsigned int); // 16.8 MB
    int* startj = (int*)(ws + off);

    int normThreads = B_ * N_ + B_ * M_;
    row_norms_kernel<<<(normThreads + 127) / 128, 128, 0, stream>>>(x, y, x2, y2);

    dim3 g(M_ / 64, N_ / 32, B_);
    sqdist_wmma_kernel<<<g, 128, 0, stream>>>(x, y, x2, y2, dmat);

    dtw_dp_kernel<<<B_, 256, 0, stream>>>(dmat, choice, startj);

    dtw_backtrack_kernel<<<B_, 128, 0, stream>>>(dmat, choice, startj, x_t, y_t, out);
}